// QuadraticFormSheafLearner_8976481648851
// MI455X (gfx1250) — compile-verified
//
#include <hip/hip_runtime.h>

typedef __attribute__((ext_vector_type(2))) float v2f;
typedef __attribute__((ext_vector_type(8))) float v8f;

#define C_DIM 64
#define P_DIM 4
#define YSTRIDE (P_DIM * C_DIM)   // 256

// ---------------------------------------------------------------------------
// Kernel 1: Y[n, p*64+c] = sum_d tensor[p,c,d] * x[n,d]
// GEMM: A = x [N x 64], B[d, n] = tensor[n*64 + d]  (n = p*64+c), D = Y [N x 256]
// One wave32 computes one 16(M) x 16(N) tile via 16x V_WMMA_F32_16X16X4_F32.
// ---------------------------------------------------------------------------
__global__ __launch_bounds__(256) void sheaf_precompute_y(
    const float* __restrict__ x,       // [N, 64]
    const float* __restrict__ tensor,  // [P, 64, 64] flat = [256, 64] as B^T
    float* __restrict__ y,             // [N, 256]
    int N, int mtiles) {
  const int wavesPerBlock = blockDim.x >> 5;
  const int wave = blockIdx.x * wavesPerBlock + (threadIdx.x >> 5);
  const int lane = threadIdx.x & 31;
  const int half = lane >> 4;   // 0: K pair {k,k+1}, 1: K pair {k+2,k+3}
  const int l    = lane & 15;

  const int totalTiles = mtiles * (YSTRIDE / 16);  // 16 n-tiles per m-tile
  if (wave >= totalTiles) return;                  // uniform per wave

  const int mtile = wave >> 4;          // consecutive waves share the A tile
  const int ntile = wave & 15;

  // A: row m, k-offset 2*half (clamp row for partial last tile; EXEC stays all-1)
  int m = mtile * 16 + l;
  int mc = (m < N) ? m : (N - 1);
  const float* __restrict__ arow = x + (size_t)mc * C_DIM + 2 * half;

  // B: column n = ntile*16 + l is contiguous in d within tensor
  const int n = ntile * 16 + l;
  const float* __restrict__ bcol = tensor + (size_t)n * C_DIM + 2 * half;

  v8f acc = {};
#pragma unroll
  for (int k = 0; k < C_DIM; k += 4) {
    v2f a = *(const v2f*)(arow + k);
    v2f b = *(const v2f*)(bcol + k);
    // (neg_a, A, neg_b, B, c_mod, C, reuse_a, reuse_b)
    acc = __builtin_amdgcn_wmma_f32_16x16x4_f32(false, a, false, b,
                                                (short)0, acc, false, false);
  }

  // D layout: VGPR i -> row (mtile*16 + i + 8*half), col (ntile*16 + l)
#pragma unroll
  for (int i = 0; i < 8; ++i) {
    int row = mtile * 16 + i + 8 * half;
    if (row < N) {
      y[(size_t)row * YSTRIDE + ntile * 16 + l] = acc[i];
    }
  }
}

// ---------------------------------------------------------------------------
// Kernel 2: one wave per edge.
//   out[e, p] = tanh( sum_c x[row[e], c] * Y[col[e], p*64 + c] )
// Lane L holds channels {2L, 2L+1}; butterfly-reduce over 32 lanes.
// ---------------------------------------------------------------------------
__global__ __launch_bounds__(256) void sheaf_edge_kernel(
    const float* __restrict__ x,   // [N, 64]
    const float* __restrict__ y,   // [N, 256]
    const int* __restrict__ ei,    // [2, E]
    float* __restrict__ out,       // [E, 4]
    int E) {
  const int wavesPerBlock = blockDim.x >> 5;
  const int e = blockIdx.x * wavesPerBlock + (threadIdx.x >> 5);
  const int lane = threadIdx.x & 31;
  if (e >= E) return;  // uniform per wave

  const int row = ei[e];       // source node
  const int col = ei[E + e];   // target node

  const float* __restrict__ xr = x + (size_t)row * C_DIM + 2 * lane;
  const float* __restrict__ yc = y + (size_t)col * YSTRIDE + 2 * lane;

  v2f xv = *(const v2f*)xr;
  v2f y0 = *(const v2f*)(yc);
  v2f y1 = *(const v2f*)(yc + 64);
  v2f y2 = *(const v2f*)(yc + 128);
  v2f y3 = *(const v2f*)(yc + 192);

  float p0 = xv.x * y0.x + xv.y * y0.y;
  float p1 = xv.x * y1.x + xv.y * y1.y;
  float p2 = xv.x * y2.x + xv.y * y2.y;
  float p3 = xv.x * y3.x + xv.y * y3.y;

#pragma unroll
  for (int off = 16; off > 0; off >>= 1) {
    p0 += __shfl_xor(p0, off, 32);
    p1 += __shfl_xor(p1, off, 32);
    p2 += __shfl_xor(p2, off, 32);
    p3 += __shfl_xor(p3, off, 32);
  }

  if (lane == 0) {
    float4 o;
    o.x = tanhf(p0);
    o.y = tanhf(p1);
    o.z = tanhf(p2);
    o.w = tanhf(p3);
    *(float4*)(out + (size_t)e * 4) = o;
  }
}

extern "C" void kernel_launch(void* const* d_in, const int* in_sizes, int n_in,
                              void* d_out, int out_size, void* d_ws, size_t ws_size,
                              hipStream_t stream) {
  const float* x      = (const float*)d_in[0];   // [N, 64]
  const int*   ei     = (const int*)d_in[1];     // [2, E]
  const float* tensor = (const float*)d_in[2];   // [4, 64, 64]
  float* out          = (float*)d_out;           // [E, 2, 2]
  float* y            = (float*)d_ws;            // [N, 256] scratch (102.4 MB)

  const int N = in_sizes[0] / C_DIM;
  const int E = in_sizes[1] / 2;

  // --- Kernel 1: precompute Y = X @ W via f32 WMMA ---
  const int mtiles = (N + 15) / 16;
  const int totalTiles = mtiles * (YSTRIDE / 16);
  const int wavesPerBlock = 8;  // 256 threads
  const int blocks1 = (totalTiles + wavesPerBlock - 1) / wavesPerBlock;
  sheaf_precompute_y<<<blocks1, 256, 0, stream>>>(x, tensor, y, N, mtiles);

  // --- Kernel 2: per-edge quadratic forms + tanh ---
  const int blocks2 = (E + wavesPerBlock - 1) / wavesPerBlock;
  sheaf_edge_kernel<<<blocks2, 256, 0, stream>>>(x, y, ei, out, E);
}